// DiffeomorphicTransform2_89593017794697
// MI455X (gfx1250) — compile-verified
//
#include <hip/hip_runtime.h>
#include <hip/hip_bf16.h>

// Diffeomorphic scaling-and-squaring (7 steps) for flow [1,3,160,192,160] f32.
// 3-D launch: lane = x (coalesced), blockIdx.y*2+ty = y, blockIdx.z = z; this
// eliminates all div/mod index math. Trilinear gathers hit the L2-resident
// field (59 MB << 192 MB L2). Ping-pong d_out <-> d_ws, final result in d_out.

namespace {
constexpr int Dd = 160;
constexpr int Hh = 192;
constexpr int Ww = 160;                       // multiple of wave32
constexpr int Nvox = Dd * Hh * Ww;            // 4,915,200
constexpr int ROWS_PER_BLOCK = 2;             // block = 160x2 = 10 waves
}

__global__ __launch_bounds__(Ww * ROWS_PER_BLOCK) void diffeo_step_kernel(
    const float* __restrict__ in, float* __restrict__ out, float scale) {
  const int x_i = threadIdx.x;                          // 0..159
  const int y_i = blockIdx.y * ROWS_PER_BLOCK + threadIdx.y;  // 0..191
  const int z_i = blockIdx.z;                           // 0..159
  const int i   = (z_i * Hh + y_i) * Ww + x_i;

  // center values (scaled — fuses the initial /2^TIME_STEP into step 1)
  const float f0 = in[i]            * scale;  // displaces z
  const float f1 = in[Nvox + i]     * scale;  // displaces y
  const float f2 = in[2 * Nvox + i] * scale;  // displaces x

  // unnormalized + clamped sample coords (align_corners=True, border padding)
  float x = fminf(fmaxf(fmaf(f2, 0.5f * (float)(Ww - 1), (float)x_i), 0.0f), (float)(Ww - 1));
  float y = fminf(fmaxf(fmaf(f1, 0.5f * (float)(Hh - 1), (float)y_i), 0.0f), (float)(Hh - 1));
  float z = fminf(fmaxf(fmaf(f0, 0.5f * (float)(Dd - 1), (float)z_i), 0.0f), (float)(Dd - 1));

  const float xf = floorf(x), yf = floorf(y), zf = floorf(z);
  const int x0 = (int)xf, y0 = (int)yf, z0 = (int)zf;
  const float wx = x - xf, wy = y - yf, wz = z - zf;
  const int x1 = min(x0 + 1, Ww - 1);
  const int y1 = min(y0 + 1, Hh - 1);
  const int z1 = min(z0 + 1, Dd - 1);

  // 8 corner linear offsets (shared by all 3 channels)
  const int b00 = (z0 * Hh + y0) * Ww;
  const int b01 = (z0 * Hh + y1) * Ww;
  const int b10 = (z1 * Hh + y0) * Ww;
  const int b11 = (z1 * Hh + y1) * Ww;
  const int i000 = b00 + x0, i001 = b00 + x1;
  const int i010 = b01 + x0, i011 = b01 + x1;
  const int i100 = b10 + x0, i101 = b10 + x1;
  const int i110 = b11 + x0, i111 = b11 + x1;

  // corner weights (computed once, reused across channels)
  const float mx = 1.0f - wx, my = 1.0f - wy, mz = 1.0f - wz;
  const float w000 = mz * my * mx, w001 = mz * my * wx;
  const float w010 = mz * wy * mx, w011 = mz * wy * wx;
  const float w100 = wz * my * mx, w101 = wz * my * wx;
  const float w110 = wz * wy * mx, w111 = wz * wy * wx;

  const float fc[3] = {f0, f1, f2};
#pragma unroll
  for (int c = 0; c < 3; ++c) {
    const float* __restrict__ p = in + c * Nvox;
    float s;
    s = w000 * p[i000];
    s = fmaf(w001, p[i001], s);
    s = fmaf(w010, p[i010], s);
    s = fmaf(w011, p[i011], s);
    s = fmaf(w100, p[i100], s);
    s = fmaf(w101, p[i101], s);
    s = fmaf(w110, p[i110], s);
    s = fmaf(w111, p[i111], s);
    // sampled field also carries the fused scale on step 1
    out[c * Nvox + i] = fmaf(scale, s, fc[c]);
  }
}

extern "C" void kernel_launch(void* const* d_in, const int* in_sizes, int n_in,
                              void* d_out, int out_size, void* d_ws, size_t ws_size,
                              hipStream_t stream) {
  (void)in_sizes; (void)n_in; (void)out_size; (void)ws_size;
  const float* flow = (const float*)d_in[0];
  float* out = (float*)d_out;
  float* ws  = (float*)d_ws;

  const dim3 block(Ww, ROWS_PER_BLOCK, 1);           // 320 threads = 10 wave32
  const dim3 grid(1, Hh / ROWS_PER_BLOCK, Dd);       // 96 x 160 blocks
  const float inv_scale = 1.0f / 128.0f;             // 2^-TIME_STEP, TIME_STEP = 7

  // Step 1 (fused initial scaling): d_in -> d_out
  diffeo_step_kernel<<<grid, block, 0, stream>>>(flow, out, inv_scale);

  // Steps 2..7 ping-pong so the final field lands in d_out:
  //   2: out->ws, 3: ws->out, 4: out->ws, 5: ws->out, 6: out->ws, 7: ws->out
  for (int s = 2; s <= 7; ++s) {
    const float* src = (s & 1) ? ws : out;
    float*       dst = (s & 1) ? out : ws;
    diffeo_step_kernel<<<grid, block, 0, stream>>>(src, dst, 1.0f);
  }
}